// Radon_4707284156587
// MI455X (gfx1250) — compile-verified
//
#include <hip/hip_runtime.h>
#include <stdint.h>

#define IMG   256          // source image width/height
#define GPAD  363          // padded grid size (256 + 107)
#define PADB  53           // pad_before
#define DEG2RAD 0.017453292519943295f

typedef uint32_t u32x4 __attribute__((ext_vector_type(4)));
typedef uint32_t u32x8 __attribute__((ext_vector_type(8)));

__device__ __forceinline__ float bilin_sample(const float* __restrict__ smem,
                                              float fi, float s, float c,
                                              float fx0, float fy0)
{
    const float fx = fmaf(fi, s, fx0);
    const float fy = fmaf(fi, c, fy0);
    const float xf = floorf(fx);
    const float yf = floorf(fy);
    float wx1 = fx - xf, wy1 = fy - yf;
    float wx0 = 1.0f - wx1, wy0 = 1.0f - wy1;

    const int xi = (int)xf - PADB;               // interior (unpadded) coords
    const int yi = (int)yf - PADB;

    // zero-padding folded into the weights (branchless)
    wx0 = ((unsigned)xi       < IMG) ? wx0 : 0.0f;
    wx1 = ((unsigned)(xi + 1) < IMG) ? wx1 : 0.0f;
    wy0 = ((unsigned)yi       < IMG) ? wy0 : 0.0f;
    wy1 = ((unsigned)(yi + 1) < IMG) ? wy1 : 0.0f;

    const int xc0 = min(max(xi,     0), IMG - 1);
    const int xc1 = min(max(xi + 1, 0), IMG - 1);
    const int yc0 = min(max(yi,     0), IMG - 1);
    const int yc1 = min(max(yi + 1, 0), IMG - 1);

    const float v00 = smem[yc0 * IMG + xc0];
    const float v01 = smem[yc0 * IMG + xc1];
    const float v10 = smem[yc1 * IMG + xc0];
    const float v11 = smem[yc1 * IMG + xc1];

    return wy0 * (wx0 * v00 + wx1 * v01)
         + wy1 * (wx0 * v10 + wx1 * v11);
}

__global__ __launch_bounds__(384)
void radon_fwd_kernel(const float* __restrict__ x,
                      const int*   __restrict__ theta,
                      float*       __restrict__ out,
                      int T)
{
    __shared__ float smem[IMG * IMG];          // 256 KB of the 320 KB WGP LDS

    const int n = blockIdx.x / T;              // image index
    const int t = blockIdx.x % T;              // angle index

    // ---------------- TDM: DMA image n (256x256 f32) global -> LDS -------------
    // One wave issues the tensor DMA (TDM ignores EXEC; branch is wave-uniform).
    if (threadIdx.x < 32) {
        const uint64_t gaddr    = (uint64_t)(const void*)(x + (size_t)n * IMG * IMG);
        const uint32_t lds_base = (uint32_t)(uintptr_t)(void*)&smem[0];

        u32x4 g0;
        g0[0] = 1u;                                  // count=1, user D#, no gather
        g0[1] = lds_base;                            // lds_addr (bytes)
        g0[2] = (uint32_t)gaddr;                     // global_addr[31:0]
        g0[3] = ((uint32_t)(gaddr >> 32) & 0x01FFFFFFu) | 0x80000000u; // addr[56:32] | type=2

        u32x8 g1;
        g1[0] = 2u << 16;                            // workgroup_mask=0, data_size=2 (4 bytes)
        g1[1] = (uint32_t)IMG << 16;                 // tensor_dim0 = 256  (bits 79:48)
        g1[2] = (uint32_t)IMG << 16;                 // tensor_dim1 = 256  (bits 111:80)
        g1[3] = (uint32_t)IMG << 16;                 // tile_dim0   = 256  (bits 127:112)
        g1[4] = (uint32_t)IMG;                       // tile_dim1 = 256, tile_dim2 = 0
        g1[5] = (uint32_t)IMG;                       // tensor_dim0_stride = 256
        g1[6] = 0u;                                  // stride0 hi = 0, stride1 lo = 0
        g1[7] = 1u;                                  // tensor_dim1_stride = 65536 (unused, 2D tile)

        asm volatile("tensor_load_to_lds %0, %1"
                     :: "s"(g0), "s"(g1)
                     : "memory");
        __builtin_amdgcn_s_wait_tensorcnt(0);        // DMA complete for this wave
    }
    __syncthreads();                                 // publish LDS to all 12 waves

    // ---------------- per-column ray march with unit steps ----------------------
    const int j = (int)threadIdx.x;
    if (j < GPAD) {
        const float th = (float)theta[t] * DEG2RAD;
        const float c  = cosf(th);
        const float s  = sinf(th);

        const float half = 0.5f * (float)(GPAD - 1);     // 181
        const float xj   = -1.0f + (float)j * (2.0f / (float)(GPAD - 1));

        // image-space sample point at i=0 (y=-1); stepping i by 1 adds (s, c)
        const float fx0 = ( c * xj - s + 1.0f) * half;
        const float fy0 = (-s * xj - c + 1.0f) * half;

        // 4 independent accumulators -> no serial add chain across the unroll
        float acc0 = 0.0f, acc1 = 0.0f, acc2 = 0.0f, acc3 = 0.0f;
        int i = 0;
        for (; i + 3 < GPAD; i += 4) {
            acc0 += bilin_sample(smem, (float)(i    ), s, c, fx0, fy0);
            acc1 += bilin_sample(smem, (float)(i + 1), s, c, fx0, fy0);
            acc2 += bilin_sample(smem, (float)(i + 2), s, c, fx0, fy0);
            acc3 += bilin_sample(smem, (float)(i + 3), s, c, fx0, fy0);
        }
        for (; i < GPAD; ++i)                          // 363 = 4*90 + 3 tail
            acc0 += bilin_sample(smem, (float)i, s, c, fx0, fy0);

        const float acc = (acc0 + acc1) + (acc2 + acc3);

        // out layout: (N, 1, GPAD, T)
        out[((size_t)n * GPAD + j) * (size_t)T + t] = acc;
    }
}

extern "C" void kernel_launch(void* const* d_in, const int* in_sizes, int n_in,
                              void* d_out, int out_size, void* d_ws, size_t ws_size,
                              hipStream_t stream) {
    const float* x     = (const float*)d_in[0];
    const int*   theta = (const int*)d_in[1];
    float*       out   = (float*)d_out;

    const int N = in_sizes[0] / (IMG * IMG);   // 2
    const int T = in_sizes[1];                 // 180

    dim3 grid((unsigned)(N * T));
    dim3 block(384);
    radon_fwd_kernel<<<grid, block, 0, stream>>>(x, theta, out, T);
}